// GNNStack_58634893525189
// MI455X (gfx1250) — compile-verified
//
#include <hip/hip_runtime.h>
#include <hip/hip_bf16.h>
#include <math.h>

// ---------------------------------------------------------------------------
// GraphSAGE stack for MI455X (gfx1250, wave32).
// Key algebraic move: relu(lin(x[src])) == relu(lin(x))[src], so the per-edge
// 128x128 GEMM collapses to a per-node GEMM + gather. Node GEMMs run on the
// WMMA f32 16x16x4 pipe (exact fp32); the edge scatter-mean is L2-resident
// atomic traffic (the true bottleneck at 23.3 TB/s / 192MB L2).
//
// Rev 2: concat K-loop split into two phases (no per-iteration cndmask),
// A fragment loaded as a single b64 (two consecutive f32, 8B-aligned).
// ---------------------------------------------------------------------------

typedef __attribute__((ext_vector_type(2))) float v2f;
typedef __attribute__((ext_vector_type(8))) float v8f;

#define NODES 10000
#define DH    128

enum { POST_NONE = 0, POST_RELU = 1, POST_RELU_NORM = 2, POST_LOGSOFTMAX = 3 };

// ---------------------------------------------------------------------------
// WMMA GEMM over a 16-row block x all M columns.
//   C[16 x M] = [A0 | A1](16 x K) @ B(K x M) + bias, then row-wise epilogue.
// One wave per 16-col tile (blockDim = 32*M/16). K stepped by 4 with
// V_WMMA_F32_16X16X4_F32. Full output rows live in one workgroup so ReLU /
// L2-normalize / log-softmax fuse here via LDS.
// ---------------------------------------------------------------------------
__device__ __forceinline__ v8f wmma_k4(const float* __restrict__ Arow,
                                       const float* __restrict__ B,
                                       int kB, int M, int colB, int half,
                                       v8f acc) {
    // A fragment (16x4 f32): lane l16 = row; half selects K pair {0,1}/{2,3}.
    // Two consecutive floats -> one 8B load.
    v2f a = *(const v2f*)(Arow + 2 * half);
    // B fragment (4x16 f32): VGPR0 = rows K+0/K+2 striped over lanes,
    // VGPR1 = rows K+1/K+3.
    v2f b = {B[(size_t)(kB + 2 * half)     * M + colB],
             B[(size_t)(kB + 2 * half + 1) * M + colB]};
    return __builtin_amdgcn_wmma_f32_16x16x4_f32(
               /*neg_a=*/false, a, /*neg_b=*/false, b,
               /*c_mod=*/(short)0, acc, /*reuse_a=*/false, /*reuse_b=*/false);
}

__global__ void sage_gemm_wmma(const float* __restrict__ A0,
                               const float* __restrict__ A1,
                               int K0, int K,
                               const float* __restrict__ B,
                               const float* __restrict__ bias,
                               float* __restrict__ C,
                               int N, int M, int post) {
    __shared__ float tile[16][DH + 4];   // padded; M <= 128
    __shared__ float rowStat[16];

    const int lane    = threadIdx.x & 31;
    const int wave    = threadIdx.x >> 5;      // column-tile index
    const int half    = lane >> 4;             // 0: lanes 0-15, 1: lanes 16-31
    const int l16     = lane & 15;
    const int rowBase = blockIdx.x * 16;
    const int colBase = wave * 16;
    const int colB    = colBase + l16;

    int rowA = rowBase + l16;
    if (rowA >= N) rowA = N - 1;               // N % 16 == 0 here; safety clamp

    v8f acc = {0.f, 0.f, 0.f, 0.f, 0.f, 0.f, 0.f, 0.f};

    // Phase 1: A0 columns [0, K0).
    {
        const float* __restrict__ Arow = A0 + (size_t)rowA * K0;
        #pragma unroll 8
        for (int k0 = 0; k0 < K0; k0 += 4)
            acc = wmma_k4(Arow + k0, B, k0, M, colB, half, acc);
    }
    // Phase 2: A1 columns [K0, K)  (concat operand; skipped when K == K0).
    if (K > K0) {
        const int ld1 = K - K0;
        const float* __restrict__ Arow = A1 + (size_t)rowA * ld1;
        #pragma unroll 8
        for (int k0 = K0; k0 < K; k0 += 4)
            acc = wmma_k4(Arow + (k0 - K0), B, k0, M, colB, half, acc);
    }

    // Bias + (ReLU if requested) while spilling C layout to LDS:
    // VGPR j holds row (j + 8*half), col (colBase + l16).
    const float bv = bias[colB];
    const bool doRelu = (post == POST_RELU) || (post == POST_RELU_NORM);
    #pragma unroll
    for (int j = 0; j < 8; ++j) {
        float v = acc[j] + bv;
        if (doRelu) v = fmaxf(v, 0.f);
        tile[j + 8 * half][colB] = v;
    }
    __syncthreads();

    // Row-wise epilogue statistics (16 rows -> threads 0..15).
    if (post == POST_RELU_NORM) {
        if (threadIdx.x < 16) {
            float ss = 0.f;
            for (int c = 0; c < M; ++c) { float v = tile[threadIdx.x][c]; ss += v * v; }
            float nrm = sqrtf(ss);
            rowStat[threadIdx.x] = 1.0f / fmaxf(nrm, 1e-12f);
        }
        __syncthreads();
    } else if (post == POST_LOGSOFTMAX) {
        if (threadIdx.x < 16) {
            float m = -INFINITY;
            for (int c = 0; c < M; ++c) m = fmaxf(m, tile[threadIdx.x][c]);
            float s = 0.f;
            for (int c = 0; c < M; ++c) s += expf(tile[threadIdx.x][c] - m);
            rowStat[threadIdx.x] = m + logf(s);
        }
        __syncthreads();
    }

    // Write back with epilogue applied.
    for (int i = threadIdx.x; i < 16 * M; i += blockDim.x) {
        const int r = i / M, col = i - r * M;
        const int grow = rowBase + r;
        if (grow >= N) continue;
        float v = tile[r][col];
        if (post == POST_RELU_NORM)       v *= rowStat[r];
        else if (post == POST_LOGSOFTMAX) v -= rowStat[r];
        C[(size_t)grow * M + col] = v;
    }
}

// ---------------------------------------------------------------------------
// Edge scatter-sum: one wave per edge. float4 gather of the transformed
// source row (L2-resident, 512B coalesced per wave), 4 float atomics per
// lane into aggr[dst], lane 0 bumps the degree counter.
// ---------------------------------------------------------------------------
__global__ void sage_scatter(const float* __restrict__ Y,
                             const long long* __restrict__ ei,   // [2, E] int64
                             float* __restrict__ aggr,
                             float* __restrict__ cnt,
                             int E) {
    const long long t = (long long)blockIdx.x * blockDim.x + threadIdx.x;
    const int e = (int)(t >> 5);
    if (e >= E) return;
    const int lane = (int)(t & 31);
    const int s = (int)ei[e];
    const int d = (int)ei[(size_t)E + e];
    const float4 v = ((const float4*)(Y + (size_t)s * DH))[lane];
    float* ar = aggr + (size_t)d * DH + lane * 4;
    atomicAdd(ar + 0, v.x);
    atomicAdd(ar + 1, v.y);
    atomicAdd(ar + 2, v.z);
    atomicAdd(ar + 3, v.w);
    if (lane == 0) atomicAdd(cnt + d, 1.0f);
}

__global__ void sage_mean(float* __restrict__ aggr, const float* __restrict__ cnt, int N) {
    const int i = blockIdx.x * blockDim.x + threadIdx.x;
    if (i >= N * DH) return;
    const int r = i >> 7;                     // DH == 128
    aggr[i] = aggr[i] / fmaxf(cnt[r], 1.0f);
}

__global__ void fill_zero(float* __restrict__ p, int n) {
    const int i = blockIdx.x * blockDim.x + threadIdx.x;
    if (i < n) p[i] = 0.f;
}

// ---------------------------------------------------------------------------
// Orchestration.
// ---------------------------------------------------------------------------
extern "C" void kernel_launch(void* const* d_in, const int* in_sizes, int n_in,
                              void* d_out, int out_size, void* d_ws, size_t ws_size,
                              hipStream_t stream) {
    const float*     x      = (const float*)d_in[0];
    const long long* ei     = (const long long*)d_in[1];
    const float*     lin_w0 = (const float*)d_in[2];
    const float*     lin_b0 = (const float*)d_in[3];
    const float*     agg_w0 = (const float*)d_in[4];
    const float*     agg_b0 = (const float*)d_in[5];
    const float*     lin_w1 = (const float*)d_in[6];
    const float*     lin_b1 = (const float*)d_in[7];
    const float*     agg_w1 = (const float*)d_in[8];
    const float*     agg_b1 = (const float*)d_in[9];
    const float*     mp_w1  = (const float*)d_in[10];
    const float*     mp_b1  = (const float*)d_in[11];
    const float*     mp_w2  = (const float*)d_in[12];
    const float*     mp_b2  = (const float*)d_in[13];
    float*           out    = (float*)d_out;

    const int N = NODES;
    const int E = in_sizes[1] / 2;   // edge_index is [2, E]

    // Workspace layout (floats): Y | AGG | CNT | H1 | H2 | T   (~25.7 MB)
    float* Y   = (float*)d_ws;
    float* AGG = Y   + (size_t)N * DH;
    float* CNT = AGG + (size_t)N * DH;       // adjacent to AGG -> one zero pass
    float* H1  = CNT + (size_t)N;
    float* H2  = H1  + (size_t)N * DH;
    float* T   = H2  + (size_t)N * DH;

    const dim3 gemmGrid(N / 16);
    const dim3 gemmBlk128(32 * (DH / 16));   // 256 threads, 8 waves
    const dim3 gemmBlk64(32 * (64 / 16));    // 128 threads, 4 waves
    const int zeroN = N * DH + N;
    const dim3 zeroGrid((zeroN + 255) / 256);
    const dim3 scatGrid((unsigned)(((long long)E * 32 + 255) / 256));
    const dim3 meanGrid((N * DH + 255) / 256);

    // ---- Layer 0 -----------------------------------------------------------
    sage_gemm_wmma<<<gemmGrid, gemmBlk128, 0, stream>>>(
        x, x, DH, DH, lin_w0, lin_b0, Y, N, DH, POST_RELU);
    fill_zero<<<zeroGrid, 256, 0, stream>>>(AGG, zeroN);
    sage_scatter<<<scatGrid, 256, 0, stream>>>(Y, ei, AGG, CNT, E);
    sage_mean<<<meanGrid, 256, 0, stream>>>(AGG, CNT, N);
    sage_gemm_wmma<<<gemmGrid, gemmBlk128, 0, stream>>>(
        x, AGG, DH, 2 * DH, agg_w0, agg_b0, H1, N, DH, POST_RELU_NORM);

    // ---- Layer 1 -----------------------------------------------------------
    sage_gemm_wmma<<<gemmGrid, gemmBlk128, 0, stream>>>(
        H1, H1, DH, DH, lin_w1, lin_b1, Y, N, DH, POST_RELU);
    fill_zero<<<zeroGrid, 256, 0, stream>>>(AGG, zeroN);
    sage_scatter<<<scatGrid, 256, 0, stream>>>(Y, ei, AGG, CNT, E);
    sage_mean<<<meanGrid, 256, 0, stream>>>(AGG, CNT, N);
    sage_gemm_wmma<<<gemmGrid, gemmBlk128, 0, stream>>>(
        H1, AGG, DH, 2 * DH, agg_w1, agg_b1, H2, N, DH, POST_RELU_NORM);

    // ---- post_mp: Linear -> Linear -> log_softmax --------------------------
    sage_gemm_wmma<<<gemmGrid, gemmBlk128, 0, stream>>>(
        H2, H2, DH, DH, mp_w1, mp_b1, T, N, DH, POST_NONE);
    sage_gemm_wmma<<<gemmGrid, gemmBlk64, 0, stream>>>(
        T, T, DH, DH, mp_w2, mp_b2, out, N, 64, POST_LOGSOFTMAX);
}